// S4DKernel_48876727828910
// MI455X (gfx1250) — compile-verified
//
#include <hip/hip_runtime.h>
#include <math.h>

typedef float v2f __attribute__((ext_vector_type(2)));
typedef float v8f __attribute__((ext_vector_type(8)));
typedef int   v4i __attribute__((ext_vector_type(4)));

#define BATCH   8
#define SEQL    4096
#define DMODEL  1024
#define NSTATE  16
#define TILE    16
#define TPC     16                 // tiles per chunk
#define CHUNKL  (TILE*TPC)         // 256
#define NCHUNK  (SEQL/CHUNKL)      // 16
#define DPB     16                 // channels per block (one per wave)

// per-channel constant-table layout (floats)
#define TBL     1408
#define O_A16R  0
#define O_A16I  16
#define O_A256R 32
#define O_A256I 48
#define O_K     64                 // 16x16 Toeplitz kernel (D folded on diag)
#define O_G     320                // 16x32 state->y correction
#define O_P     832                // 32x16 u->local-state

#define WS_SLOC ((size_t)DMODEL*TBL)                       // chunk-local end states
#define STATE_F ((size_t)DMODEL*NCHUNK*BATCH*NSTATE*2)
#define WS_HIN  (WS_SLOC + STATE_F)                        // chunk incoming states

// LDS tile: [row(16 t)][col(16 b,cc)][dloc padded to 20] -> 16B-aligned float4
// slots AND bank-conflict-free column reads (20*c mod 64 distinct for c=0..15).
#define DPAD    20
#define TILEF   (16*16*DPAD)       // 5120 floats / 20480 B

// ---- CDNA5 async global->LDS path (guarded; falls back to sync ld/st) ------
#if defined(__has_builtin)
#if __has_builtin(__builtin_amdgcn_global_load_async_to_lds_b128)
#define USE_ASYNC 1
#endif
#endif
#ifndef USE_ASYNC
#define USE_ASYNC 0
#endif

#if defined(__has_builtin)
#if __has_builtin(__builtin_amdgcn_s_wait_asynccnt)
#define WAIT_ASYNC(n) __builtin_amdgcn_s_wait_asynccnt(n)   // n must be an ICE
#endif
#endif
#ifndef WAIT_ASYNC
#if USE_ASYNC
#define WAIT_ASYNC(n) asm volatile("s_wait_asynccnt %0" ::"i"(n))
#else
#define WAIT_ASYNC(n) ((void)0)
#endif
#endif

__device__ __forceinline__ size_t state_addr(int d, int c, int b, int n) {
  return ((((size_t)d * NCHUNK + c) * BATCH + b) * NSTATE + n) * 2;
}

// One K=4 slice of a 16x16xK f32 matmul on the matrix core.
__device__ __forceinline__ v8f wmma4(float a0, float a1, float b0, float b1, v8f c) {
  v2f a; a[0] = a0; a[1] = a1;
  v2f b; b[0] = b0; b[1] = b1;
  return __builtin_amdgcn_wmma_f32_16x16x4_f32(false, a, false, b, (short)0, c,
                                               false, false);
}

// Cooperative (512-thread) fetch of one u tile into LDS.
// Tile element (row,col,dloc): row = t within tile, col = b + 8*cc, dloc = d-dbase.
// 1024 float4 slots, 2 per thread; global segment = 16 consecutive channels.
__device__ __forceinline__ void issue_tile_load(
    const float* __restrict__ u, float* __restrict__ lbuf,
    int dbase, int cp, int trel, int tid)
{
  #pragma unroll
  for (int k = 0; k < 2; ++k) {
    int slot = tid + 512 * k;
    int row  = slot >> 6;
    int col  = (slot >> 2) & 15;
    int q    = slot & 3;
    int b    = col & 7;
    int c    = 2 * cp + (col >> 3);
    const float* g = u + ((size_t)b * SEQL + (size_t)c * CHUNKL + trel + row) * DMODEL
                       + dbase + 4 * q;
    float* l = lbuf + (row * 16 + col) * DPAD + 4 * q;
#if USE_ASYNC
    float* gm = const_cast<float*>(g);
    __builtin_amdgcn_global_load_async_to_lds_b128(
        (__attribute__((address_space(1))) v4i*)gm,
        (__attribute__((address_space(3))) v4i*)l, 0, 0);
#else
    const float4 val = *(const float4*)g;
    *(float4*)l = val;
#endif
  }
}

// ---------------------------------------------------------------------------
// Kernel 1: per-channel tables.
// ---------------------------------------------------------------------------
__global__ __launch_bounds__(256) void s4d_precompute(
    const float* __restrict__ Bm, const float* __restrict__ Cm,
    const float* __restrict__ log_dt, const float* __restrict__ Dv,
    const float* __restrict__ logAr, const float* __restrict__ Aim,
    float* __restrict__ ws)
{
  int d = blockIdx.x * blockDim.x + threadIdx.x;
  if (d >= DMODEL) return;
  float* tab = ws + (size_t)d * TBL;
  float dt = fminf(fmaxf(expf(log_dt[d]), 1e-4f), 1.0f);

  float kacc[16];
  #pragma unroll
  for (int t = 0; t < 16; ++t) kacc[t] = 0.f;

  for (int n = 0; n < NSTATE; ++n) {
    float er  = expf(logAr[n]);
    float mag = expf(-er * dt);
    float th  = Aim[n] * dt;
    float abr = mag * cosf(th);
    float abi = mag * sinf(th);
    float Bb  = dt * Bm[d * NSTATE + n];
    float Cn  = Cm[d * NSTATE + n];
    float pr = 1.f, pi = 0.f;                 // Abar^0
    for (int t = 0; t < 16; ++t) {
      kacc[t] += Cn * Bb * pr;
      tab[O_P + (2*n)   * 16 + (15 - t)] = Bb * pr;
      tab[O_P + (2*n+1) * 16 + (15 - t)] = Bb * pi;
      float nr = pr * abr - pi * abi;
      float ni = pr * abi + pi * abr;
      pr = nr; pi = ni;                       // Abar^{t+1}
      tab[O_G + t * 32 + 2*n]     =  Cn * pr;
      tab[O_G + t * 32 + 2*n + 1] = -Cn * pi;
    }
    tab[O_A16R + n] = pr; tab[O_A16I + n] = pi;
    float qr = pr, qi = pi;
    #pragma unroll
    for (int s = 0; s < 4; ++s) { float tr = qr*qr - qi*qi; qi = 2.f*qr*qi; qr = tr; }
    tab[O_A256R + n] = qr; tab[O_A256I + n] = qi;
  }
  float Dd = Dv[d];
  for (int t = 0; t < 16; ++t)
    for (int s = 0; s < 16; ++s)
      tab[O_K + t * 16 + s] = (t >= s) ? (kacc[t - s] + (t == s ? Dd : 0.f)) : 0.f;
}

// D-layout (two 16x16 state tiles) -> B-operand layout; half-swap via shfl_xor(16).
__device__ __forceinline__ void rearrange_states(const v8f LT, const v8f LB,
                                                 bool hiLane, float* out16)
{
  #pragma unroll
  for (int j = 0; j < 8; ++j) {
    #pragma unroll
    for (int v = 0; v < 2; ++v) {
      int  jm = j & 3;
      float lo, hi;
      if (jm < 2) {
        float dlow = (j < 4) ? LT[4*jm + v]     : LB[4*jm + v];
        float src  = (j < 4) ? LT[4*jm + v + 2] : LB[4*jm + v + 2];
        lo = dlow;
        hi = __shfl_xor(src, 16, 32);
      } else {
        float src  = (j < 4) ? LT[4*(jm-2) + v]     : LB[4*(jm-2) + v];
        float dhi  = (j < 4) ? LT[4*(jm-2) + v + 2] : LB[4*(jm-2) + v + 2];
        lo = __shfl_xor(src, 16, 32);
        hi = dhi;
      }
      out16[2*j + v] = hiLane ? hi : lo;
    }
  }
}

// ---------------------------------------------------------------------------
// Kernel 2: chunk-local end states.  Block = 16 waves = 16 channels; wave =
// one d x 16 columns (8 batches x 2 chunks).  u staged via async->LDS.
// ---------------------------------------------------------------------------
__global__ __launch_bounds__(512) void s4d_chunk_states(
    const float* __restrict__ u, float* __restrict__ ws)
{
  __shared__ float ubuf[2][TILEF];

  const int  tid    = threadIdx.x;
  const int  w      = tid >> 5;             // wave id == local channel
  const int  lane   = tid & 31;
  const int  dbase  = blockIdx.x * DPB;
  const int  d      = dbase + w;
  const int  cp     = blockIdx.y;           // chunk pair
  const bool hiLane = lane >= 16;
  const int  o2     = hiLane ? 2 : 0;
  const int  l15    = lane & 15;
  const int  b      = l15 & 7;
  const int  c      = 2 * cp + (l15 >> 3);
  const float* tab  = ws + (size_t)d * TBL;
  float* sloc       = ws + WS_SLOC;

  float pT[8], pB[8];
  #pragma unroll
  for (int jj = 0; jj < 4; ++jj)
    #pragma unroll
    for (int v = 0; v < 2; ++v) {
      pT[2*jj + v] = tab[O_P + l15        * 16 + 4*jj + v + o2];
      pB[2*jj + v] = tab[O_P + (16 + l15) * 16 + 4*jj + v + o2];
    }
  float arr[8], ari[8];
  #pragma unroll
  for (int j = 0; j < 8; ++j) {
    int n = 2*j + (hiLane ? 1 : 0);
    arr[j] = tab[O_A16R + n]; ari[j] = tab[O_A16I + n];
  }

  float hb[16];
  #pragma unroll
  for (int i = 0; i < 16; ++i) hb[i] = 0.f;

  issue_tile_load(u, ubuf[0], dbase, cp, 0, tid);

  for (int tile = 0; tile < TPC; ++tile) {
    if (tile + 1 < TPC) {
      issue_tile_load(u, ubuf[(tile + 1) & 1], dbase, cp, (tile + 1) * TILE, tid);
      WAIT_ASYNC(2);                         // tile's 2 loads done, next 2 in flight
    } else {
      WAIT_ASYNC(0);
    }
    __syncthreads();                         // tile data visible block-wide

    const float* ub = ubuf[tile & 1];
    float uB[8];
    #pragma unroll
    for (int jj = 0; jj < 4; ++jj)
      #pragma unroll
      for (int v = 0; v < 2; ++v)
        uB[2*jj + v] = ub[((4*jj + v + o2) * 16 + l15) * DPAD + w];

    v8f LT = {0,0,0,0,0,0,0,0}, LB = {0,0,0,0,0,0,0,0};
    #pragma unroll
    for (int jj = 0; jj < 4; ++jj) {
      LT = wmma4(pT[2*jj], pT[2*jj+1], uB[2*jj], uB[2*jj+1], LT);
      LB = wmma4(pB[2*jj], pB[2*jj+1], uB[2*jj], uB[2*jj+1], LB);
    }
    float add[16];
    rearrange_states(LT, LB, hiLane, add);
    #pragma unroll
    for (int j = 0; j < 8; ++j) {
      float hr = hb[2*j], hi_ = hb[2*j+1];
      hb[2*j]   = arr[j]*hr - ari[j]*hi_ + add[2*j];
      hb[2*j+1] = ari[j]*hr + arr[j]*hi_ + add[2*j+1];
    }
    __syncthreads();                         // protect buffer reuse
  }

  #pragma unroll
  for (int j = 0; j < 8; ++j) {
    int n = 2*j + (hiLane ? 1 : 0);
    size_t a = state_addr(d, c, b, n);
    sloc[a]     = hb[2*j];
    sloc[a + 1] = hb[2*j+1];
  }
}

// ---------------------------------------------------------------------------
// Kernel 3: inter-chunk combine.  h_in[c+1] = A^256 . h_in[c] + local[c].
// ---------------------------------------------------------------------------
__global__ __launch_bounds__(256) void s4d_combine(float* __restrict__ ws)
{
  int tid = blockIdx.x * blockDim.x + threadIdx.x;   // DMODEL*BATCH*NSTATE
  int d = tid >> 7;
  int b = (tid >> 4) & 7;
  int n = tid & 15;
  const float* tab  = ws + (size_t)d * TBL;
  const float* sloc = ws + WS_SLOC;
  float*       hin  = ws + WS_HIN;
  float ar = tab[O_A256R + n], ai = tab[O_A256I + n];
  float hr = 0.f, hi = 0.f;
  for (int c = 0; c < NCHUNK; ++c) {
    size_t a = state_addr(d, c, b, n);
    hin[a] = hr; hin[a + 1] = hi;
    float lr = sloc[a], li = sloc[a + 1];
    float nr = ar*hr - ai*hi + lr;
    float ni = ai*hr + ar*hi + li;
    hr = nr; hi = ni;
  }
}

// ---------------------------------------------------------------------------
// Kernel 4: outputs.  Y = K@U + G@h_in (12 WMMAs), y transposed through LDS
// for coalesced float4 stores; carry via P@U (8 WMMAs) + diag A^16.
// ---------------------------------------------------------------------------
__global__ __launch_bounds__(512) void s4d_output(
    const float* __restrict__ u, const float* __restrict__ ws,
    float* __restrict__ y)
{
  __shared__ float ubuf[2][TILEF];
  __shared__ float ybuf[TILEF];

  const int  tid    = threadIdx.x;
  const int  w      = tid >> 5;
  const int  lane   = tid & 31;
  const int  dbase  = blockIdx.x * DPB;
  const int  d      = dbase + w;
  const int  cp     = blockIdx.y;
  const bool hiLane = lane >= 16;
  const int  o2     = hiLane ? 2 : 0;
  const int  l15    = lane & 15;
  const int  b      = l15 & 7;
  const int  c      = 2 * cp + (l15 >> 3);
  const float* tab  = ws + (size_t)d * TBL;
  const float* hin  = ws + WS_HIN;

  float kA[8], gA[16], pT[8], pB[8];
  #pragma unroll
  for (int jj = 0; jj < 4; ++jj)
    #pragma unroll
    for (int v = 0; v < 2; ++v) {
      kA[2*jj + v] = tab[O_K + l15 * 16 + 4*jj + v + o2];
      pT[2*jj + v] = tab[O_P + l15        * 16 + 4*jj + v + o2];
      pB[2*jj + v] = tab[O_P + (16 + l15) * 16 + 4*jj + v + o2];
    }
  #pragma unroll
  for (int jj = 0; jj < 8; ++jj)
    #pragma unroll
    for (int v = 0; v < 2; ++v)
      gA[2*jj + v] = tab[O_G + l15 * 32 + 4*jj + v + o2];

  float arr[8], ari[8];
  #pragma unroll
  for (int j = 0; j < 8; ++j) {
    int n = 2*j + (hiLane ? 1 : 0);
    arr[j] = tab[O_A16R + n]; ari[j] = tab[O_A16I + n];
  }

  float hb[16];
  #pragma unroll
  for (int j = 0; j < 8; ++j) {
    int n = 2*j + (hiLane ? 1 : 0);
    size_t a = state_addr(d, c, b, n);
    hb[2*j]   = hin[a];
    hb[2*j+1] = hin[a + 1];
  }

  issue_tile_load(u, ubuf[0], dbase, cp, 0, tid);

  for (int tile = 0; tile < TPC; ++tile) {
    if (tile + 1 < TPC) {
      issue_tile_load(u, ubuf[(tile + 1) & 1], dbase, cp, (tile + 1) * TILE, tid);
      WAIT_ASYNC(2);
    } else {
      WAIT_ASYNC(0);
    }
    __syncthreads();

    const float* ub = ubuf[tile & 1];
    float uB[8];
    #pragma unroll
    for (int jj = 0; jj < 4; ++jj)
      #pragma unroll
      for (int v = 0; v < 2; ++v)
        uB[2*jj + v] = ub[((4*jj + v + o2) * 16 + l15) * DPAD + w];

    // Y = K@U + G@h_in
    v8f acc = {0,0,0,0,0,0,0,0};
    #pragma unroll
    for (int jj = 0; jj < 4; ++jj)
      acc = wmma4(kA[2*jj], kA[2*jj+1], uB[2*jj], uB[2*jj+1], acc);
    #pragma unroll
    for (int jj = 0; jj < 8; ++jj)
      acc = wmma4(gA[2*jj], gA[2*jj+1], hb[2*jj], hb[2*jj+1], acc);

    // stage y tile in LDS (transpose to d-contiguous), then coalesced store
    #pragma unroll
    for (int r = 0; r < 8; ++r)
      ybuf[((r + (hiLane ? 8 : 0)) * 16 + l15) * DPAD + w] = acc[r];
    __syncthreads();

    #pragma unroll
    for (int k = 0; k < 2; ++k) {
      int slot = tid + 512 * k;
      int row  = slot >> 6;
      int col  = (slot >> 2) & 15;
      int q    = slot & 3;
      int sb   = col & 7;
      int sc   = 2 * cp + (col >> 3);
      float4 val = *(const float4*)&ybuf[(row * 16 + col) * DPAD + 4 * q];
      *(float4*)(y + ((size_t)sb * SEQL + (size_t)sc * CHUNKL + tile * TILE + row) * DMODEL
                   + dbase + 4 * q) = val;
    }

    // carry: h = A^16 . h + P@U
    v8f LT = {0,0,0,0,0,0,0,0}, LB = {0,0,0,0,0,0,0,0};
    #pragma unroll
    for (int jj = 0; jj < 4; ++jj) {
      LT = wmma4(pT[2*jj], pT[2*jj+1], uB[2*jj], uB[2*jj+1], LT);
      LB = wmma4(pB[2*jj], pB[2*jj+1], uB[2*jj], uB[2*jj+1], LB);
    }
    float add[16];
    rearrange_states(LT, LB, hiLane, add);
    #pragma unroll
    for (int j = 0; j < 8; ++j) {
      float hr = hb[2*j], hi_ = hb[2*j+1];
      hb[2*j]   = arr[j]*hr - ari[j]*hi_ + add[2*j];
      hb[2*j+1] = ari[j]*hr + arr[j]*hi_ + add[2*j+1];
    }
    __syncthreads();                         // ybuf + ubuf reuse fence
  }
}

extern "C" void kernel_launch(void* const* d_in, const int* in_sizes, int n_in,
                              void* d_out, int out_size, void* d_ws, size_t ws_size,
                              hipStream_t stream) {
  (void)in_sizes; (void)n_in; (void)out_size; (void)ws_size;
  const float* u      = (const float*)d_in[0];
  const float* Bm     = (const float*)d_in[1];
  const float* Cm     = (const float*)d_in[2];
  const float* log_dt = (const float*)d_in[3];
  const float* Dv     = (const float*)d_in[4];
  const float* logAr  = (const float*)d_in[5];
  const float* Aim    = (const float*)d_in[6];
  float* y  = (float*)d_out;
  float* ws = (float*)d_ws;

  dim3 grid2(DMODEL / DPB, NCHUNK / 2);      // 64 x 8 blocks, 512 threads
  s4d_precompute<<<DMODEL / 256, 256, 0, stream>>>(Bm, Cm, log_dt, Dv, logAr, Aim, ws);
  s4d_chunk_states<<<grid2, 512, 0, stream>>>(u, ws);
  s4d_combine<<<(DMODEL * BATCH * NSTATE) / 256, 256, 0, stream>>>(ws);
  s4d_output<<<grid2, 512, 0, stream>>>(u, ws, y);
}